// GaussianScene2_28063316312551
// MI455X (gfx1250) — compile-verified
//
#include <hip/hip_runtime.h>
#include <hip/hip_bf16.h>
#include <stdint.h>

// ---------------------------------------------------------------------------
// Gaussian-splat preprocess, MI455X (gfx1250).
// Roofline: 144 B/point * 2M = 288 MB -> ~12.4 us at 23.3 TB/s; ~0.5 GFLOP of
// ALU is noise against that, so the kernel is engineered around data movement:
//   * double-buffered GLOBAL_LOAD_ASYNC_TO_LDS_B128 staging (ASYNCcnt +
//     s_wait_asynccnt) turns the 12-byte-stride AoS inputs into full-cacheline
//     streams and keeps tile k+1 DMA in flight under tile k compute;
//   * all outputs are streamed once -> non-temporal stores keep the 192MB L2
//     available for the input stream.
// The reference's one GEMM-shaped op, pc = [p,1] @ extrinsic ((Nx4)@(4x4)), is
// mapped onto V_WMMA_F32_16X16X4_F32: A is built per-lane straight from the
// LDS-staged points (ISA 16x4 layout needs no cross-lane shuffles), B is the
// loop-invariant extrinsic. The D-matrix (column-per-lane) is un-transposed
// through a wave-private LDS scratch (wave-local DS ops are in-order, no
// barrier). This costs ~nothing at the 12.4us BW bound but runs the matrix op
// on the matrix pipe. The remaining 3x3 covariance algebra stays in VALU:
// its lane-layout transposes would cost more than the ALU they replace.
// ---------------------------------------------------------------------------

typedef float f4  __attribute__((ext_vector_type(4)));
typedef float v2f __attribute__((ext_vector_type(2)));
typedef float v8f __attribute__((ext_vector_type(8)));

#if defined(__gfx1250__) && __has_builtin(__builtin_amdgcn_wmma_f32_16x16x4_f32)
#define HAVE_WMMA4 1
#else
#define HAVE_WMMA4 0
#endif

#define TILE 256
// Byte offsets of the staged arrays inside one LDS buffer (all 16B aligned).
#define OFF_PTS 0u        // 256 * 3 * 4 = 3072 B
#define OFF_QTS 3072u     // 256 * 4 * 4 = 4096 B
#define OFF_SCL 7168u     // 3072 B
#define OFF_COL 10240u    // 3072 B
#define OFF_OPA 13312u    // 1024 B
#define BUF_BYTES 14336u
#define BUF_FLOATS 3584   // 14336 / 4

__device__ __forceinline__ void async_cp16(uint32_t ldsAddr, uint64_t gAddr) {
  // One 16B global->LDS async copy per lane (tracked by ASYNCcnt).
  asm volatile("global_load_async_to_lds_b128 %0, %1, off"
               :: "v"(ldsAddr), "v"(gAddr)
               : "memory");
}

// Issues EXACTLY 5 async instructions per lane (wave-uniform control flow) so
// ASYNCcnt bookkeeping is a compile-time constant. Lanes beyond the needed
// chunk count redundantly re-copy the last chunk (benign identical-data race).
__device__ __forceinline__ void stage_tile(
    const char* pts, const char* qts, const char* scl,
    const char* col, const char* opa,
    uint32_t ldsBase, long long p0, int L, int tid)
{
  int need, c;
  need = (L * 12 + 15) >> 4; c = tid < need ? tid : need - 1;            // points
  async_cp16(ldsBase + OFF_PTS + (uint32_t)c * 16u,
             (uint64_t)(uintptr_t)(pts + p0 * 12 + (long long)c * 16));
  need = L;                  c = tid < need ? tid : need - 1;            // quats
  async_cp16(ldsBase + OFF_QTS + (uint32_t)c * 16u,
             (uint64_t)(uintptr_t)(qts + p0 * 16 + (long long)c * 16));
  need = (L * 12 + 15) >> 4; c = tid < need ? tid : need - 1;            // scales
  async_cp16(ldsBase + OFF_SCL + (uint32_t)c * 16u,
             (uint64_t)(uintptr_t)(scl + p0 * 12 + (long long)c * 16));
  async_cp16(ldsBase + OFF_COL + (uint32_t)c * 16u,                      // colors
             (uint64_t)(uintptr_t)(col + p0 * 12 + (long long)c * 16));
  need = (L * 4 + 15) >> 4;  c = tid < need ? tid : need - 1;            // opacity
  async_cp16(ldsBase + OFF_OPA + (uint32_t)c * 16u,
             (uint64_t)(uintptr_t)(opa + p0 * 4 + (long long)c * 16));
}

__device__ __forceinline__ int clampi(int v, int lo, int hi) {
  v = v < lo ? lo : v;
  return v > hi ? hi : v;
}

__global__ void __launch_bounds__(TILE)
gs_forward_kernel(const float* __restrict__ points,
                  const float* __restrict__ quats,
                  const float* __restrict__ scales,
                  const float* __restrict__ colors,
                  const float* __restrict__ opacity,
                  const float* __restrict__ E,      // extrinsic 4x4 row-major
                  const float* __restrict__ I,      // intrinsic 4x4 row-major
                  const int* __restrict__ p_fx,
                  const int* __restrict__ p_fy,
                  const int* __restrict__ p_w,
                  const int* __restrict__ p_h,
                  const int* __restrict__ p_ts,
                  float* __restrict__ out,
                  int n)
{
  __shared__ float smem[2][BUF_FLOATS];
  __shared__ f4 scratch4[8 * 32];            // 4KB wave-private D un-transpose
  float* scratch = (float*)scratch4;

  const int tid = threadIdx.x;
  const long long nLL = (long long)n;
  const int numTiles = (n + TILE - 1) / TILE;

  // Wave-uniform scalars (compiler emits s_load for these).
  const float fx  = (float)p_fx[0];
  const float fy  = (float)p_fy[0];
  const float fw  = (float)p_w[0];
  const float fh  = (float)p_h[0];
  const float tsF = (float)p_ts[0];
  // tan(2*atan(w/(2f))/2) == w/(2f) exactly.
  const float tanX = fw / (2.0f * fx);
  const float tanY = fh / (2.0f * fy);
  const int max_tx = (int)ceilf(fw / tsF) - 1;
  const int max_ty = (int)ceilf(fh / tsF) - 1;

  float Em[16], Im[16];
#pragma unroll
  for (int k = 0; k < 16; ++k) { Em[k] = E[k]; Im[k] = I[k]; }

  // Low 32 bits of a generic LDS pointer are the LDS byte offset (flat->LDS
  // aperture mapping truncates to addr[31:0]).
  const uint32_t ldsBase0 = (uint32_t)(uintptr_t)(&smem[0][0]);

  const char* ptsB = (const char*)points;
  const char* qtsB = (const char*)quats;
  const char* sclB = (const char*)scales;
  const char* colB = (const char*)colors;
  const char* opaB = (const char*)opacity;

#if HAVE_WMMA4
  // Loop-invariant WMMA operands / lane roles.
  const int  lane  = tid & 31;
  const int  wv    = tid >> 5;
  const int  colid = lane & 15;
  const bool lolane = lane < 16;
  float* wscr = scratch + wv * 128;          // this wave's 32-point pc slab
  // B (4x16): col n<4 holds extrinsic column n, cols 4..15 are zero.
  // ISA layout: lanes 0-15 carry K=0,1 in v0,v1; lanes 16-31 carry K=2,3.
  v2f Bm; Bm.x = 0.0f; Bm.y = 0.0f;
  if (colid < 4) {
    Bm.x = lolane ? Em[colid]     : Em[8 + colid];
    Bm.y = lolane ? Em[4 + colid] : Em[12 + colid];
  }
#endif

  const int t0 = blockIdx.x;
  const int stride = gridDim.x;

  // Prime buffer 0 with this block's first tile.
  {
    long long p0 = (long long)t0 * TILE;
    long long rem = nLL - p0;
    int L = rem < TILE ? (int)rem : TILE;
    stage_tile(ptsB, qtsB, sclB, colB, opaB, ldsBase0, p0, L, tid);
  }

  int buf = 0;
  for (int t = t0; t < numTiles; t += stride) {
    const int tn = t + stride;
    if (tn < numTiles) {
      // Prefetch next tile into the other buffer, then wait only for the 5
      // older copies (current buffer) -> next tile's DMA overlaps compute.
      long long p0n = (long long)tn * TILE;
      long long remn = nLL - p0n;
      int Ln = remn < TILE ? (int)remn : TILE;
      stage_tile(ptsB, qtsB, sclB, colB, opaB,
                 ldsBase0 + (uint32_t)(buf ^ 1) * BUF_BYTES, p0n, Ln, tid);
      asm volatile("s_wait_asynccnt 5" ::: "memory");
    } else {
      asm volatile("s_wait_asynccnt 0" ::: "memory");
    }
    __syncthreads();

    const long long p0 = (long long)t * TILE;
    const long long idx = p0 + tid;
    const float* S = smem[buf];

#if HAVE_WMMA4
    // pc = [p,1] @ extrinsic for this wave's 32 points via two
    // V_WMMA_F32_16X16X4_F32 (full EXEC here: uniform control flow, 256-thread
    // blocks). A layout (16x4 f32): lane L<16 -> row L, K={x,y};
    // lane L>=16 -> row L-16, K={z,1}. Built straight from staged LDS.
    float pc0, pc1, pc2, pc3;
    {
      const float* pb0 = S + OFF_PTS / 4 + ((wv << 5) + colid) * 3;
      const float* pb1 = pb0 + 48;           // +16 points * 3 floats
      v2f A0, A1;
      A0.x = lolane ? pb0[0] : pb0[2];
      A0.y = lolane ? pb0[1] : 1.0f;
      A1.x = lolane ? pb1[0] : pb1[2];
      A1.y = lolane ? pb1[1] : 1.0f;
      v8f cz = (v8f)0.0f;
      v8f d0 = __builtin_amdgcn_wmma_f32_16x16x4_f32(
          false, A0, false, Bm, (short)0, cz, false, false);
      v8f d1 = __builtin_amdgcn_wmma_f32_16x16x4_f32(
          false, A1, false, Bm, (short)0, cz, false, false);
      // D layout: lane n<16 = col n rows 0-7 (v0..7), lane n+16 = rows 8-15.
      // Only cols 0-3 are live; un-transpose via wave-private LDS slab
      // (wave-local DS ops are in-order -> no barrier needed).
      if (colid < 4) {
        const int rb = lolane ? 0 : 8;
#pragma unroll
        for (int r = 0; r < 8; ++r) {
          wscr[(rb + r) * 4 + colid]      = d0[r];
          wscr[64 + (rb + r) * 4 + colid] = d1[r];
        }
      }
      const f4 pcv = *(const f4*)(wscr + lane * 4);
      pc0 = pcv.x; pc1 = pcv.y; pc2 = pcv.z; pc3 = pcv.w;
    }
#endif

    if (idx < nLL) {
      const float* Qp = S + OFF_QTS / 4 + tid * 4;
      float qr = Qp[0], qx = Qp[1], qy = Qp[2], qz = Qp[3];
      const float* Sp = S + OFF_SCL / 4 + tid * 3;
      const float  s0 = Sp[0], s1 = Sp[1], s2 = Sp[2];
      const float* Cp = S + OFF_COL / 4 + tid * 3;
      const float  col0 = Cp[0], col1 = Cp[1], col2 = Cp[2];
      const float  opac = S[OFF_OPA / 4 + tid];

#if !HAVE_WMMA4
      const float* Pp = S + OFF_PTS / 4 + tid * 3;
      const float wx = Pp[0], wy = Pp[1], wz = Pp[2];
      const float pc0 = wx * Em[0] + wy * Em[4] + wz * Em[8]  + Em[12];
      const float pc1 = wx * Em[1] + wy * Em[5] + wz * Em[9]  + Em[13];
      const float pc2 = wx * Em[2] + wy * Em[6] + wz * Em[10] + Em[14];
      const float pc3 = wx * Em[3] + wy * Em[7] + wz * Em[11] + Em[15];
#endif

      // Quaternion -> rotation
      const float qn = sqrtf(qr * qr + qx * qx + qy * qy + qz * qz);
      qr /= qn; qx /= qn; qy /= qn; qz /= qn;
      const float xx = qx * qx, yy = qy * qy, zz = qz * qz;
      const float xy = qx * qy, xz = qx * qz, yz = qy * qz;
      const float rx = qr * qx, ry = qr * qy, rz = qr * qz;
      const float R00 = 1.0f - 2.0f * (yy + zz), R01 = 2.0f * (xy - rz), R02 = 2.0f * (xz + ry);
      const float R10 = 2.0f * (xy + rz), R11 = 1.0f - 2.0f * (xx + zz), R12 = 2.0f * (yz - rx);
      const float R20 = 2.0f * (xz - ry), R21 = 2.0f * (yz + rx), R22 = 1.0f - 2.0f * (xx + yy);

      // M = R * diag-broadcast(scales); cov3d = M M^T (symmetric)
      const float M00 = R00 * s0, M01 = R01 * s1, M02 = R02 * s2;
      const float M10 = R10 * s0, M11 = R11 * s1, M12 = R12 * s2;
      const float M20 = R20 * s0, M21 = R21 * s1, M22 = R22 * s2;
      const float cv00 = M00 * M00 + M01 * M01 + M02 * M02;
      const float cv01 = M00 * M10 + M01 * M11 + M02 * M12;
      const float cv02 = M00 * M20 + M01 * M21 + M02 * M22;
      const float cv11 = M10 * M10 + M11 * M11 + M12 * M12;
      const float cv12 = M10 * M20 + M11 * M21 + M12 * M22;
      const float cv22 = M20 * M20 + M21 * M21 + M22 * M22;

      const float zc = pc2;
      const bool  zmask = zc > 0.2f;
      const float zs = zmask ? zc : 1.0f;
      const float xcl = fminf(fmaxf(pc0 / zs, -1.3f * tanX), 1.3f * tanX) * zc;
      const float ycl = fminf(fmaxf(pc1 / zs, -1.3f * tanY), 1.3f * tanY) * zc;
      const float zs2 = zs * zs;
      const float j00 = fx / zs;
      const float j02 = -(fx * xcl) / zs2;
      const float j11 = fy / zs;
      const float j12 = -(fy * ycl) / zs2;

      // T = Wm @ J^T ; u = T[:,0], v = T[:,1]
      const float u0 = Em[0] * j00 + Em[2]  * j02;
      const float u1 = Em[4] * j00 + Em[6]  * j02;
      const float u2 = Em[8] * j00 + Em[10] * j02;
      const float v0 = Em[1] * j11 + Em[2]  * j12;
      const float v1 = Em[5] * j11 + Em[6]  * j12;
      const float v2 = Em[9] * j11 + Em[10] * j12;

      // C = T^T cov3d T (only 2x2 block needed)
      const float cu0 = cv00 * u0 + cv01 * u1 + cv02 * u2;
      const float cu1 = cv01 * u0 + cv11 * u1 + cv12 * u2;
      const float cu2 = cv02 * u0 + cv12 * u1 + cv22 * u2;
      const float cw0 = cv00 * v0 + cv01 * v1 + cv02 * v2;
      const float cw1 = cv01 * v0 + cv11 * v1 + cv12 * v2;
      const float cw2 = cv02 * v0 + cv12 * v1 + cv22 * v2;
      const float ca = (u0 * cu0 + u1 * cu1 + u2 * cu2) + 0.3f; // C00+0.3
      const float cb =  u0 * cw0 + u1 * cw1 + u2 * cw2;         // C01
      const float cc =  v0 * cu0 + v1 * cu1 + v2 * cu2;         // C10
      const float cd = (v0 * cw0 + v1 * cw1 + v2 * cw2) + 0.3f; // C11+0.3

      // ndc = pc @ intrinsic (kept in VALU to preserve the reference's
      // two-step association)
      const float ndc0 = pc0 * Im[0] + pc1 * Im[4] + pc2 * Im[8]  + pc3 * Im[12];
      const float ndc1 = pc0 * Im[1] + pc1 * Im[5] + pc2 * Im[9]  + pc3 * Im[13];
      const float ndc2 = pc0 * Im[2] + pc1 * Im[6] + pc2 * Im[10] + pc3 * Im[14];
      const float ndc3 = pc0 * Im[3] + pc1 * Im[7] + pc2 * Im[11] + pc3 * Im[15];
      const float ws = zmask ? ndc3 : 1.0f;
      const float nx = ndc0 / ws;
      const float ny = ndc1 / ws;
      const float nz = ndc2;
      const bool mask = (nz > 0.2f) && (nx < 1.3f) && (nx > -1.3f)
                                    && (ny < 1.3f) && (ny > -1.3f);

      const float det = ca * cd - cb * cc;
      const float dets = (fabsf(det) < 1e-12f) ? 1e-12f : det;
      const float i00 =  cd / dets;
      const float i01 = -cb / dets;
      const float i10 = -cc / dets;
      const float i11 =  ca / dets;
      const float mid = 0.5f * (ca + cd);
      const float sq = sqrtf(fmaxf(mid * mid - det, 0.1f));
      const float lmax = fmaxf(mid + sq, mid - sq);
      const float rad = ceilf(3.0f * sqrtf(fmaxf(lmax, 1e-6f)));
      const float pxs = ((nx + 1.0f) * fw - 1.0f) * 0.5f;
      const float pys = ((ny + 1.0f) * fh - 1.0f) * 0.5f;

      const int tlx = clampi((int)((pxs - rad) / tsF), 0, max_tx);
      const int tly = clampi((int)((pys - rad) / tsF), 0, max_ty);
      const int brx = clampi((int)((pxs + rad) / tsF), 0, max_tx);
      const int bry = clampi((int)((pys + rad) / tsF), 0, max_ty);
      int sx = brx + 1 - tlx; if (sx < 1) sx = 1;
      int sy = bry + 1 - tly; if (sy < 1) sy = 1;
      const int tiles = mask ? sx * sy : 0;

      // ---- streamed (non-temporal) outputs ----
      const f4 z4 = (f4)0.0f;
      f4 o0 = mask ? (f4){pxs, pys, nz, ca} : z4;
      f4 o1 = mask ? (f4){cb, cc, cd, i00} : z4;
      f4 o2 = mask ? (f4){i01, i10, i11, rad} : z4;
      f4 o3 = mask ? (f4){col0, col1, col2, opac} : z4;
      f4* orow = (f4*)(out + idx * 16);
      __builtin_nontemporal_store(o0, orow + 0);
      __builtin_nontemporal_store(o1, orow + 1);
      __builtin_nontemporal_store(o2, orow + 2);
      __builtin_nontemporal_store(o3, orow + 3);

      float* oT  = out + 16LL * nLL;
      float* oTL = out + 17LL * nLL;
      float* oBR = out + 19LL * nLL;
      float* oM  = out + 21LL * nLL;
      __builtin_nontemporal_store((float)tiles, oT + idx);
      __builtin_nontemporal_store((float)tlx, oTL + idx);
      __builtin_nontemporal_store((float)tly, oTL + nLL + idx);
      __builtin_nontemporal_store((float)brx, oBR + idx);
      __builtin_nontemporal_store((float)bry, oBR + nLL + idx);
      __builtin_nontemporal_store(mask ? 1.0f : 0.0f, oM + idx);
    }
    __syncthreads();   // all reads of `buf` done before it is restaged
    buf ^= 1;
  }
}

extern "C" void kernel_launch(void* const* d_in, const int* in_sizes, int n_in,
                              void* d_out, int out_size, void* d_ws, size_t ws_size,
                              hipStream_t stream) {
  (void)n_in; (void)out_size; (void)d_ws; (void)ws_size;
  const float* points  = (const float*)d_in[0];
  const float* quats   = (const float*)d_in[1];
  const float* scales  = (const float*)d_in[2];
  const float* colors  = (const float*)d_in[3];
  const float* opacity = (const float*)d_in[4];
  const float* extr    = (const float*)d_in[5];
  const float* intr    = (const float*)d_in[6];
  const int*   fx      = (const int*)d_in[7];
  const int*   fy      = (const int*)d_in[8];
  const int*   w       = (const int*)d_in[9];
  const int*   h       = (const int*)d_in[10];
  const int*   ts      = (const int*)d_in[11];

  const int n = in_sizes[0] / 3;
  if (n <= 0) return;
  const int numTiles = (n + TILE - 1) / TILE;
  int grid = (numTiles + 3) / 4;   // ~4 tiles per block => double-buffer pays
  if (grid < 1) grid = 1;
  if (grid > numTiles) grid = numTiles;

  gs_forward_kernel<<<grid, TILE, 0, stream>>>(
      points, quats, scales, colors, opacity, extr, intr,
      fx, fy, w, h, ts, (float*)d_out, n);
}